// MixedReservoir_61753039781958
// MI455X (gfx1250) — compile-verified
//
#include <hip/hip_runtime.h>
#include <math.h>

typedef __attribute__((ext_vector_type(2))) float v2f;
typedef __attribute__((ext_vector_type(8))) float v8f;

#define D      2048
#define TB     2048      // T*B sequential steps
#define BATCH  16
#define TT     128
#define NOUT   256
#define NWG    64        // scan grid size (persistent, grid-barriered)
#define SCTH   256       // scan block threads

// workspace layout (float element offsets)
#define WS_U0  0
#define WS_U1  (WS_U0 + TB*D)
#define WS_H1  (WS_U1 + TB*D)
#define WS_S0  (WS_H1 + TB*D)
#define WS_S1  (WS_S0 + D)
#define WS_CTR (WS_S1 + D)

// ---------------------------------------------------------------------------
// init: copy initial states into ws, reset grid-barrier counter
// ---------------------------------------------------------------------------
__global__ __launch_bounds__(256) void mr_init_kernel(const float* __restrict__ s0i,
                                                      const float* __restrict__ s1i,
                                                      float* __restrict__ ws) {
  int i = blockIdx.x * blockDim.x + threadIdx.x;
  if (i < D) {
    ws[WS_S0 + i] = s0i[i];
    ws[WS_S1 + i] = s1i[i];
  }
  if (i == 0) *(unsigned int*)(ws + WS_CTR) = 0u;
}

// ---------------------------------------------------------------------------
// U = X @ trans{0,1}, fp32 WMMA 16x16x4. Rows are in xs order (t-major):
// xs_row r maps to x[(r%16)*128 + r/16]. Each wave: 64x16 output strip
// (4 M-tiles sharing one B fragment per k-step).
// grid: (Mstrips=32, Nblocks=32, mat=2), block = 128 (4 waves, wave = ntile)
// ---------------------------------------------------------------------------
__global__ __launch_bounds__(128) void mr_gemm_u_kernel(const float* __restrict__ x,
                                                        const float* __restrict__ trans0,
                                                        const float* __restrict__ trans1,
                                                        float* __restrict__ ws) {
  const int wave = threadIdx.x >> 5;
  const int lane = threadIdx.x & 31;
  const int half = lane >> 4;       // 0: k,k+1  1: k+2,k+3
  const int l16  = lane & 15;

  const int mstrip = blockIdx.x;                 // 64 rows each
  const int ntile  = blockIdx.y * 4 + wave;      // 0..127
  const int mat    = blockIdx.z;                 // 0 or 1

  const float* __restrict__ W = mat ? trans1 : trans0;
  float* __restrict__ U = ws + (mat ? WS_U1 : WS_U0);

  const int n = ntile * 16 + l16;

  // per-lane base offsets into x for the 4 A tiles (row = xs row)
  int rowbase[4];
#pragma unroll
  for (int i = 0; i < 4; ++i) {
    int r = mstrip * 64 + i * 16 + l16;
    rowbase[i] = ((r & 15) * TT + (r >> 4)) * D;   // b*T + t row of x
  }

  v8f c[4] = {v8f{}, v8f{}, v8f{}, v8f{}};

  for (int kk = 0; kk < D; kk += 4) {
    const int kb = kk + 2 * half;
    v2f b;
    b.x = W[(size_t)kb * D + n];
    b.y = W[(size_t)(kb + 1) * D + n];
    v2f a0 = *(const v2f*)(x + rowbase[0] + kb);
    v2f a1 = *(const v2f*)(x + rowbase[1] + kb);
    v2f a2 = *(const v2f*)(x + rowbase[2] + kb);
    v2f a3 = *(const v2f*)(x + rowbase[3] + kb);
    c[0] = __builtin_amdgcn_wmma_f32_16x16x4_f32(false, a0, false, b, (short)0, c[0], false, false);
    c[1] = __builtin_amdgcn_wmma_f32_16x16x4_f32(false, a1, false, b, (short)0, c[1], false, false);
    c[2] = __builtin_amdgcn_wmma_f32_16x16x4_f32(false, a2, false, b, (short)0, c[2], false, false);
    c[3] = __builtin_amdgcn_wmma_f32_16x16x4_f32(false, a3, false, b, (short)0, c[3], false, false);
  }

#pragma unroll
  for (int i = 0; i < 4; ++i) {
#pragma unroll
    for (int r = 0; r < 8; ++r) {
      int m = mstrip * 64 + i * 16 + r + 8 * half;   // C layout: VGPR r -> M=r / r+8
      U[(size_t)m * D + n] = c[i][r];
    }
  }
}

// ---------------------------------------------------------------------------
// sequential reservoir scan: persistent kernel, NWG workgroups, grid barrier
// per step via atomic counter. Each WG owns 32 output columns per reservoir.
// ---------------------------------------------------------------------------
__device__ __forceinline__ float mr_nl(float c, int col) {
  if (col < 1024) return sinf(c);
  if (col < 1536) return tanhf(c);
  return logf(c * c);
}

__global__ __launch_bounds__(SCTH) void mr_scan_kernel(const float* __restrict__ res0,
                                                       const float* __restrict__ res1,
                                                       const float* __restrict__ alpha0,
                                                       const float* __restrict__ alpha1,
                                                       float* __restrict__ ws) {
  __shared__ float v0s[D];
  __shared__ float v1s[D];
  __shared__ float red[SCTH];

  const int tid = threadIdx.x;
  const int wg  = blockIdx.x;
  const int out_local = tid & 63;     // 0..63 (32 cols x 2 reservoirs)
  const int chunk = tid >> 6;         // 0..3 k-chunks of 512
  const int resv  = out_local >> 5;   // 0 or 1
  const int col   = wg * 32 + (out_local & 31);

  const float* __restrict__ U0 = ws + WS_U0;
  const float* __restrict__ U1 = ws + WS_U1;
  float* __restrict__ H1 = ws + WS_H1;
  float* __restrict__ s0 = ws + WS_S0;
  float* __restrict__ s1 = ws + WS_S1;
  unsigned int* ctr = (unsigned int*)(ws + WS_CTR);

  const float* __restrict__ R  = resv ? res1 : res0;
  const float alpha = (resv ? alpha1 : alpha0)[col];
  float* __restrict__ sv = resv ? s1 : s0;

  for (int st = 0; st < TB; ++st) {
    // stage v = U[st] + s into LDS (both reservoirs)
    for (int i = tid; i < D; i += SCTH) {
      v0s[i] = U0[(size_t)st * D + i] + s0[i];
      v1s[i] = U1[(size_t)st * D + i] + s1[i];
    }
    __syncthreads();

    const float* vs = resv ? v1s : v0s;
    float acc = 0.0f;
    const int k0 = chunk * (D / 4);
#pragma unroll 4
    for (int k = k0; k < k0 + D / 4; ++k) {
      acc = fmaf(vs[k], R[(size_t)k * D + col], acc);
    }
    red[tid] = acc;
    __syncthreads();

    if (tid < 64) {
      float s = red[tid] + red[tid + 64] + red[tid + 128] + red[tid + 192];
      float h = alpha * mr_nl(s, col);
      sv[col] = h;                          // new state
      if (resv) H1[(size_t)st * D + col] = h;  // readout input
    }
    __threadfence();
    __syncthreads();

    // grid-wide barrier: monotonically increasing counter
    if (tid == 0) {
      __hip_atomic_fetch_add(ctr, 1u, __ATOMIC_ACQ_REL, __HIP_MEMORY_SCOPE_AGENT);
      const unsigned int target = (unsigned int)(NWG) * (unsigned int)(st + 1);
      while (__hip_atomic_load(ctr, __ATOMIC_ACQUIRE, __HIP_MEMORY_SCOPE_AGENT) < target) {
        __builtin_amdgcn_s_sleep(1);
      }
    }
    __syncthreads();
  }
}

// ---------------------------------------------------------------------------
// readout: Y = H1 @ lin_w^T + lin_b, fp32 WMMA, permuted store [T,B]->[B,T]
// grid: (Mstrips=32, Nblocks=4), block = 128 (4 waves, wave = ntile 0..15)
// ---------------------------------------------------------------------------
__global__ __launch_bounds__(128) void mr_readout_kernel(const float* __restrict__ lin_w,
                                                         const float* __restrict__ lin_b,
                                                         const float* __restrict__ ws,
                                                         float* __restrict__ out) {
  const int wave = threadIdx.x >> 5;
  const int lane = threadIdx.x & 31;
  const int half = lane >> 4;
  const int l16  = lane & 15;

  const int mstrip = blockIdx.x;               // 64 rows each
  const int ntile  = blockIdx.y * 4 + wave;    // 0..15
  const int n = ntile * 16 + l16;

  const float* __restrict__ H1 = ws + WS_H1;

  v8f c[4] = {v8f{}, v8f{}, v8f{}, v8f{}};

  for (int kk = 0; kk < D; kk += 4) {
    const int kb = kk + 2 * half;
    v2f b;                                      // B[k][n] = lin_w[n*D + k]
    b.x = lin_w[(size_t)n * D + kb];
    b.y = lin_w[(size_t)n * D + kb + 1];
    v2f a0 = *(const v2f*)(H1 + (size_t)(mstrip * 64 + 0 * 16 + l16) * D + kb);
    v2f a1 = *(const v2f*)(H1 + (size_t)(mstrip * 64 + 1 * 16 + l16) * D + kb);
    v2f a2 = *(const v2f*)(H1 + (size_t)(mstrip * 64 + 2 * 16 + l16) * D + kb);
    v2f a3 = *(const v2f*)(H1 + (size_t)(mstrip * 64 + 3 * 16 + l16) * D + kb);
    c[0] = __builtin_amdgcn_wmma_f32_16x16x4_f32(false, a0, false, b, (short)0, c[0], false, false);
    c[1] = __builtin_amdgcn_wmma_f32_16x16x4_f32(false, a1, false, b, (short)0, c[1], false, false);
    c[2] = __builtin_amdgcn_wmma_f32_16x16x4_f32(false, a2, false, b, (short)0, c[2], false, false);
    c[3] = __builtin_amdgcn_wmma_f32_16x16x4_f32(false, a3, false, b, (short)0, c[3], false, false);
  }

  const float bias = lin_b[n];
#pragma unroll
  for (int i = 0; i < 4; ++i) {
#pragma unroll
    for (int r = 0; r < 8; ++r) {
      int row = mstrip * 64 + i * 16 + r + 8 * half;  // xs step index
      int bb = row & 15;                              // batch
      int tt = row >> 4;                              // time
      out[((size_t)bb * TT + tt) * NOUT + n] = c[i][r] + bias;
    }
  }
}

// ---------------------------------------------------------------------------
extern "C" void kernel_launch(void* const* d_in, const int* in_sizes, int n_in,
                              void* d_out, int out_size, void* d_ws, size_t ws_size,
                              hipStream_t stream) {
  (void)in_sizes; (void)n_in; (void)out_size; (void)ws_size;

  const float* x      = (const float*)d_in[0];
  const float* trans0 = (const float*)d_in[1];
  const float* trans1 = (const float*)d_in[2];
  const float* res0   = (const float*)d_in[3];
  const float* res1   = (const float*)d_in[4];
  const float* alpha0 = (const float*)d_in[5];
  const float* alpha1 = (const float*)d_in[6];
  const float* lin_w  = (const float*)d_in[7];
  const float* lin_b  = (const float*)d_in[8];
  const float* state0 = (const float*)d_in[9];
  const float* state1 = (const float*)d_in[10];
  float* out = (float*)d_out;
  float* ws  = (float*)d_ws;

  mr_init_kernel<<<dim3(8), dim3(256), 0, stream>>>(state0, state1, ws);
  mr_gemm_u_kernel<<<dim3(32, 32, 2), dim3(128), 0, stream>>>(x, trans0, trans1, ws);
  mr_scan_kernel<<<dim3(NWG), dim3(SCTH), 0, stream>>>(res0, res1, alpha0, alpha1, ws);
  mr_readout_kernel<<<dim3(32, 4), dim3(128), 0, stream>>>(lin_w, lin_b, ws, out);
}